// SessionPop_26508538151782
// MI455X (gfx1250) — compile-verified
//
#include <hip/hip_runtime.h>
#include <stdint.h>

#define NWAVES 8
#define THREADS 256
#define CHUNK 256  // elements per wave-chunk (1KB ids + 1KB ones)

// d_out layout:
//   mode 0 (out_size == 2*V): [ idx : V x int32 ][ cnt : V x f32 ]   (JAX x64-off default)
//   mode 1 (out_size == 3*V): [ idx : V x int64 ][ cnt : V x f32 ]
__global__ void init_out_kernel(float* __restrict__ out, long long V, int mode) {
  long long i = (long long)blockIdx.x * blockDim.x + threadIdx.x;
  if (i >= V) return;
  if (mode == 0) {
    ((int*)out)[i] = (int)i;
    out[V + i] = 0.0f;
  } else {
    ((long long*)out)[i] = i;
    out[2 * V + i] = 0.0f;
  }
}

__global__ __launch_bounds__(THREADS) void hist_async_kernel(
    const int* __restrict__ ids, const float* __restrict__ ones,
    float* __restrict__ cnt, long long n, long long V) {
  // Per-wave private double buffers: no cross-wave barriers needed.
  __shared__ int   s_ids[2][NWAVES][CHUNK];
  __shared__ float s_ones[2][NWAVES][CHUNK];

  const unsigned lane = threadIdx.x & 31u;
  const unsigned wv   = threadIdx.x >> 5;
  const long long wave    = (long long)blockIdx.x * NWAVES + wv;
  const long long wstride = (long long)gridDim.x * NWAVES;
  const long long nChunks = (n + CHUNK - 1) / CHUNK;

  const unsigned long long idsBase  = (unsigned long long)(uintptr_t)ids;
  const unsigned long long onesBase = (unsigned long long)(uintptr_t)ones;
  const unsigned long long nBytes   = (unsigned long long)n * 4ull;

  // Issue one chunk's async DMA: 2KB per wave = 4 x b128 per lane (ids+ones interleaved).
  auto issue = [&](long long c, int buf) {
    const unsigned ldsI = (unsigned)(uintptr_t)&s_ids[buf][wv][0];   // low 32 bits = LDS offset
    const unsigned ldsO = (unsigned)(uintptr_t)&s_ones[buf][wv][0];
    const unsigned long long cOff = (unsigned long long)c * (CHUNK * 4ull);
#pragma unroll
    for (int k = 0; k < 2; ++k) {
      const unsigned long long seg = (unsigned long long)(k * 32 + lane) * 16ull;
      // Clamp fully-OOB 16B segments to the buffer base (junk ignored at process time).
      unsigned long long gi = idsBase  + cOff + seg;
      unsigned long long go = onesBase + cOff + seg;
      gi = (cOff + seg >= nBytes) ? idsBase  : gi;
      go = (cOff + seg >= nBytes) ? onesBase : go;
      asm volatile("global_load_async_to_lds_b128 %0, %1, off"
                   :: "v"(ldsI + (unsigned)seg), "v"(gi) : "memory");
      asm volatile("global_load_async_to_lds_b128 %0, %1, off"
                   :: "v"(ldsO + (unsigned)seg), "v"(go) : "memory");
    }
  };

  long long c = wave;
  int cur = 0;
  if (c < nChunks) issue(c, 0);
  while (c < nChunks) {
    const long long cn = c + wstride;
    const bool hasNext = (cn < nChunks);
    if (hasNext) {
      issue(cn, cur ^ 1);                                  // 4 ops in flight for next
      asm volatile("s_wait_asynccnt 4" ::: "memory");      // current chunk's 4 done
    } else {
      asm volatile("s_wait_asynccnt 0" ::: "memory");      // drain
    }
    const long long base = c * CHUNK;
#pragma unroll
    for (int j = 0; j < CHUNK / 32; ++j) {
      const long long g = base + j * 32 + lane;            // stride-32 LDS reads: conflict-free
      const int   id = s_ids[cur][wv][j * 32 + lane];
      const float w  = s_ones[cur][wv][j * 32 + lane];
      if (g < n && (unsigned long long)(unsigned)id < (unsigned long long)V) {
        const unsigned long long addr = (unsigned long long)(uintptr_t)(cnt + id);
        // Non-returning native f32 atomic add (TH=0); lands in L2 (4MB cnt << 192MB L2).
        asm volatile("global_atomic_add_f32 %0, %1, off" :: "v"(addr), "v"(w) : "memory");
      }
    }
    c = cn;
    cur ^= 1;
  }
}

extern "C" void kernel_launch(void* const* d_in, const int* in_sizes, int n_in,
                              void* d_out, int out_size, void* d_ws, size_t ws_size,
                              hipStream_t stream) {
  (void)n_in; (void)d_ws; (void)ws_size;
  const long long n = (long long)in_sizes[0];

  long long V; int mode;
  if (out_size % 2 == 0) { V = out_size / 2; mode = 0; }   // int32 idx + f32 cnt
  else                   { V = out_size / 3; mode = 1; }   // int64 idx + f32 cnt
  float* out = (float*)d_out;
  float* cnt = out + (mode == 0 ? V : 2 * V);

  // Re-zero cnt and rewrite idx every call (atomics accumulate; harness replays).
  const int tb = 256;
  const unsigned initBlocks = (unsigned)((V + tb - 1) / tb);
  hipLaunchKernelGGL(init_out_kernel, dim3(initBlocks), dim3(tb), 0, stream, out, V, mode);

  const long long nChunks = (n + CHUNK - 1) / CHUNK;       // 51200 for 13.1M elems
  long long waves  = (nChunks + 3) / 4;                    // ~4 chunks per wave
  long long blocks = (waves + NWAVES - 1) / NWAVES;
  if (blocks < 1) blocks = 1;
  if (blocks > 4096) blocks = 4096;
  hipLaunchKernelGGL(hist_async_kernel, dim3((unsigned)blocks), dim3(THREADS), 0, stream,
                     (const int*)d_in[0], (const float*)d_in[1], cnt, n, V);
}